// Circle_Loss_24086176596035
// MI455X (gfx1250) — compile-verified
//
#include <hip/hip_runtime.h>

typedef __attribute__((ext_vector_type(2))) float v2f;
typedef __attribute__((ext_vector_type(4))) float v4f;
typedef __attribute__((ext_vector_type(8))) float v8f;

#define N_DIM    8192u
#define N_QUADS  (N_DIM * N_DIM / 4u)   /* 16,777,216 float4 chunks */
#define GAMMA    64.0f
#define MARG     0.25f

// wave32 butterfly reduction (CDNA5 is wave32-only)
static __device__ __forceinline__ float waveReduceAdd(float v) {
#pragma unroll
  for (int off = 16; off > 0; off >>= 1)
    v += __shfl_xor(v, off, 32);
  return v;
}

__global__ void __launch_bounds__(64) zero_acc_kernel(float* __restrict__ acc) {
  if (threadIdx.x < 2) acc[threadIdx.x] = 0.0f;
}

// Streams all 64M entries once (NT b128 loads), computes exp-logits, masks the
// diagonal, and folds into the WMMA f32 C accumulator using B = ones.
// D = A x ones + C  ==> every C column accumulates row-sums of A.
__global__ void __launch_bounds__(256) sum_neg_kernel(
    const float* __restrict__ scores, float* __restrict__ acc) {
  const unsigned tid      = blockIdx.x * 256u + threadIdx.x;
  const unsigned nthreads = gridDim.x * 256u;      // 524,288 -> exactly 32 iters
  const v2f ones = {1.0f, 1.0f};
  v8f c = {0.f, 0.f, 0.f, 0.f, 0.f, 0.f, 0.f, 0.f};

  for (unsigned i = tid; i < N_QUADS; i += nthreads) {
    const unsigned base = i * 4u;                  // flat element index
    const unsigned row  = base >> 13;              // / 8192
    const unsigned col  = base & (N_DIM - 1u);     // % 8192 (multiple of 4)

    const v4f s = __builtin_nontemporal_load((const v4f*)scores + i);

    float e[4];
#pragma unroll
    for (int k = 0; k < 4; ++k) {
      const float sv = s[k];
      const float an = fmaxf(sv + MARG, 0.0f);
      const float lg = an * (sv - MARG) * GAMMA;
      const float ev = __expf(lg);
      e[k] = (row == col + (unsigned)k) ? 0.0f : ev;  // zero the diagonal
    }

    const v2f a0 = {e[0], e[1]};
    const v2f a1 = {e[2], e[3]};
    // v_wmma_f32_16x16x4_f32: 8-arg form (neg_a, A, neg_b, B, c_mod, C, reuse_a, reuse_b)
    c = __builtin_amdgcn_wmma_f32_16x16x4_f32(false, a0, false, ones, (short)0, c, false, false);
    c = __builtin_amdgcn_wmma_f32_16x16x4_f32(false, a1, false, ones, (short)0, c, false, false);
  }

  // Sum of all C entries == 16 * (actual sum); the /16 happens in finalize.
  float s = ((c[0] + c[1]) + (c[2] + c[3])) + ((c[4] + c[5]) + (c[6] + c[7]));
  s = waveReduceAdd(s);

  __shared__ float lds[8];
  const int lane = threadIdx.x & 31;
  const int wid  = threadIdx.x >> 5;
  if (lane == 0) lds[wid] = s;
  __syncthreads();
  if (threadIdx.x == 0) {
    float t = 0.0f;
#pragma unroll
    for (int w = 0; w < 8; ++w) t += lds[w];
    atomicAdd(&acc[0], t);                         // global_atomic_add_f32
  }
}

// Diagonal term: 8192 strided loads, single workgroup, negligible cost.
__global__ void __launch_bounds__(256) sum_pos_kernel(
    const float* __restrict__ scores, float* __restrict__ acc) {
  float s = 0.0f;
  for (unsigned i = threadIdx.x; i < N_DIM; i += 256u) {
    const float sp = scores[(size_t)i * (size_t)(N_DIM + 1u)];
    const float ap = fmaxf((1.0f + MARG) - sp, 0.0f);
    const float lp = -ap * (sp - (1.0f - MARG)) * GAMMA;
    s += __expf(lp);
  }
  s = waveReduceAdd(s);

  __shared__ float lds[8];
  const int lane = threadIdx.x & 31;
  const int wid  = threadIdx.x >> 5;
  if (lane == 0) lds[wid] = s;
  __syncthreads();
  if (threadIdx.x == 0) {
    float t = 0.0f;
#pragma unroll
    for (int w = 0; w < 8; ++w) t += lds[w];
    acc[1] = t;                                    // single block: plain store
  }
}

__global__ void finalize_kernel(const float* __restrict__ acc,
                                float* __restrict__ out) {
  const float sum_n = acc[0] * (1.0f / 16.0f);     // undo 16x column replication
  const float sum_p = acc[1];
  float prod = sum_n * sum_p;                      // inf propagates, clamped next
  prod = fminf(prod, 1e38f);
  out[0] = logf(1.0f + prod);
}

extern "C" void kernel_launch(void* const* d_in, const int* in_sizes, int n_in,
                              void* d_out, int out_size, void* d_ws, size_t ws_size,
                              hipStream_t stream) {
  const float* scores = (const float*)d_in[0];
  float* out = (float*)d_out;
  float* acc = (float*)d_ws;   // acc[0] = 16*sum_n, acc[1] = sum_p

  zero_acc_kernel<<<1, 64, 0, stream>>>(acc);
  // 2048 blocks x 256 threads = 524,288 threads; 16,777,216 quads / 524,288
  // = exactly 32 iterations per thread -> uniform EXEC (WMMA requirement).
  sum_neg_kernel<<<2048, 256, 0, stream>>>(scores, acc);
  sum_pos_kernel<<<1, 256, 0, stream>>>(scores, acc);
  finalize_kernel<<<1, 1, 0, stream>>>(acc, out);
}